// TridiagonalSolver_51170240364846
// MI455X (gfx1250) — compile-verified
//
#include <hip/hip_runtime.h>
#include <hip/hip_bf16.h>

// Batched tridiagonal solve (Thomas-equivalent) for NX=8192 systems of size
// NV=2048, f32. Memory-bound: 256 MiB in + 64 MiB out => ~14us at 23.3 TB/s.
// Strategy: one workgroup per system, full system staged in LDS via CDNA5
// async global->LDS copies (ASYNCcnt-tracked, b128 coalesced), solved with
// in-place cyclic reduction (exact direct solver, same solution as Thomas),
// written back with async LDS->global stores. No scratch traffic.
// LDS indices are padded (i -> i + i/32) so CR's power-of-two strides hit
// distinct banks on the 64-bank LDS instead of serializing.

#ifndef __has_builtin
#define __has_builtin(x) 0
#endif

#define NV 2048
#define BS 256
#define LEVELS 11                // log2(NV)
#define NVP (NV + NV / 32)       // padded LDS extent: 2112

#define AS1 __attribute__((address_space(1)))
#define AS3 __attribute__((address_space(3)))

typedef int v4i_t __attribute__((__vector_size__(16)));

// Bank-conflict-avoiding index map: one pad DWORD per 32 elements.
// 16B (4-float) chunks never straddle a pad (32 % 4 == 0), so b128 async
// copies to/from &arr[P(e)] with e % 4 == 0 stay contiguous.
__device__ __forceinline__ int P(int i) { return i + (i >> 5); }

// ---- CDNA5 async global<->LDS helpers (ASYNCcnt-tracked) -------------------

__device__ __forceinline__ void async_load_b128(const float* g, float* l) {
#if __has_builtin(__builtin_amdgcn_global_load_async_to_lds_b128)
    __builtin_amdgcn_global_load_async_to_lds_b128(
        (AS1 v4i_t*)(g), (AS3 v4i_t*)(l), /*offset=*/0, /*cpol=*/0);
#else
    unsigned loff = (unsigned)(unsigned long long)(AS3 void*)(l);
    asm volatile("global_load_async_to_lds_b128 %0, %1, off"
                 :: "v"(loff), "v"(g) : "memory");
#endif
}

__device__ __forceinline__ void async_store_b128(float* g, const float* l) {
#if __has_builtin(__builtin_amdgcn_global_store_async_from_lds_b128)
    __builtin_amdgcn_global_store_async_from_lds_b128(
        (AS1 v4i_t*)(g), (AS3 v4i_t*)(l), /*offset=*/0, /*cpol=*/0);
#else
    unsigned loff = (unsigned)(unsigned long long)(AS3 void*)(l);
    asm volatile("global_store_async_from_lds_b128 %0, %1, off"
                 :: "v"(loff), "v"(g) : "memory");
#endif
}

__device__ __forceinline__ void async_wait_all() {
#if __has_builtin(__builtin_amdgcn_s_wait_asynccnt)
    __builtin_amdgcn_s_wait_asynccnt(0);
#else
    asm volatile("s_wait_asynccnt 0" ::: "memory");
#endif
}

// Fast reciprocal: v_rcp_f32 + one Newton-Raphson step (~full f32 accuracy,
// avoids the IEEE divide expansion; system is diagonally dominant).
__device__ __forceinline__ float fast_rcp(float x) {
    float r = __builtin_amdgcn_rcpf(x);
    r = r * (2.0f - x * r);
    return r;
}

// ---- kernel: one block per system ------------------------------------------

__global__ __launch_bounds__(BS)
void TridiagonalSolver_51170240364846_kernel(const float* __restrict__ ga,
                                             const float* __restrict__ gb,
                                             const float* __restrict__ gc,
                                             const float* __restrict__ gd,
                                             float* __restrict__ gout) {
    __shared__ float sa[NVP];
    __shared__ float sb[NVP];
    __shared__ float sc[NVP];
    __shared__ float sd[NVP];  // becomes x during back-substitution

    const int tid = threadIdx.x;
    const size_t base = (size_t)blockIdx.x * NV;

    // ---- stage the whole system into LDS with async b128 copies (coalesced)
    #pragma unroll
    for (int j = 0; j < NV / 4; j += BS) {
        const int e = (j + tid) * 4;  // float index, 16B aligned
        const int p = P(e);
        async_load_b128(ga + base + e, &sa[p]);
        async_load_b128(gb + base + e, &sb[p]);
        async_load_b128(gc + base + e, &sc[p]);
        async_load_b128(gd + base + e, &sd[p]);
    }
    async_wait_all();
    __syncthreads();

    // ---- forward cyclic reduction -----------------------------------------
    // After level `lev`, surviving equations sit at i == 2^(lev+1)-1 (mod
    // 2^(lev+1)) and couple unknowns i +/- 2^(lev+1). x beyond [0,NV) is 0
    // (matches the reference's zero-initialized back-substitution).
    for (int lev = 0; lev < LEVELS; ++lev) {
        const int stride = 1 << lev;
        const int cnt = NV >> (lev + 1);
        for (int k = tid; k < cnt; k += BS) {
            const int i  = (2 * k + 2) * stride - 1;
            const int pi = P(i);
            const int pL = P(i - stride);
            const int iR = i + stride;

            const float ai = sa[pi];
            const float ci = sc[pi];

            const float alpha = -ai * fast_rcp(sb[pL]);
            float bi = sb[pi] + alpha * sc[pL];
            float di = sd[pi] + alpha * sd[pL];
            const float na = alpha * sa[pL];
            float nc;
            if (iR < NV) {
                const int pR = P(iR);
                const float gamma = -ci * fast_rcp(sb[pR]);
                bi += gamma * sa[pR];
                di += gamma * sd[pR];
                nc = gamma * sc[pR];
            } else {
                nc = 0.0f;  // x[>=NV] == 0
            }
            sa[pi] = na;
            sb[pi] = bi;
            sc[pi] = nc;
            sd[pi] = di;
        }
        __syncthreads();
    }

    // ---- apex: single remaining equation at i = NV-1 ----------------------
    if (tid == 0) {
        const int p = P(NV - 1);
        sd[p] = sd[p] * fast_rcp(sb[p]);
    }
    __syncthreads();

    // ---- back-substitution (x written in place into sd) -------------------
    for (int lev = LEVELS - 1; lev >= 0; --lev) {
        const int stride = 1 << lev;
        const int cnt = NV >> (lev + 1);
        for (int k = tid; k < cnt; k += BS) {
            const int i  = (2 * k + 1) * stride - 1;
            const int pi = P(i);
            const int iL = i - stride;
            const float xL = (iL >= 0) ? sd[P(iL)] : 0.0f;  // x[-1] == 0
            const float xR = sd[P(i + stride)];
            sd[pi] = (sd[pi] - sa[pi] * xL - sc[pi] * xR) * fast_rcp(sb[pi]);
        }
        __syncthreads();
    }

    // ---- write solution: async LDS->global, coalesced b128 ----------------
    #pragma unroll
    for (int j = 0; j < NV / 4; j += BS) {
        const int e = (j + tid) * 4;
        async_store_b128(gout + base + e, &sd[P(e)]);
    }
    async_wait_all();
}

// ---- host-side launch ------------------------------------------------------

extern "C" void kernel_launch(void* const* d_in, const int* in_sizes, int n_in,
                              void* d_out, int out_size, void* d_ws, size_t ws_size,
                              hipStream_t stream) {
    (void)n_in; (void)out_size; (void)d_ws; (void)ws_size;
    const float* a = (const float*)d_in[0];
    const float* b = (const float*)d_in[1];
    const float* c = (const float*)d_in[2];
    const float* d = (const float*)d_in[3];
    float* out = (float*)d_out;

    const int nx = in_sizes[0] / NV;  // 8192 independent systems
    dim3 grid(nx), block(BS);
    hipLaunchKernelGGL(TridiagonalSolver_51170240364846_kernel,
                       grid, block, 0, stream, a, b, c, d, out);
}